// GCBlock_30013231464710
// MI455X (gfx1250) — compile-verified
//
#include <hip/hip_runtime.h>
#include <hip/hip_bf16.h>
#include <math.h>

#define BB 2
#define CC 256
#define CHC 128
#define NN 32768   // 32*32*32
#define LN_EPS 1e-5f

typedef float v2f __attribute__((ext_vector_type(2)));
typedef float v8f __attribute__((ext_vector_type(8)));

// ---------------------------------------------------------------------------
// Kernel 1: mask[st][b][n] = sum_c x[b][c][n]*wm[c] + bm   (both S and T)
// ---------------------------------------------------------------------------
__global__ void k_mask(const float* __restrict__ S, const float* __restrict__ T,
                       const float* __restrict__ wm_s, const float* __restrict__ bm_s,
                       const float* __restrict__ wm_t, const float* __restrict__ bm_t,
                       float* __restrict__ mask) {
    const int n = blockIdx.x * blockDim.x + threadIdx.x;
    const int b = blockIdx.y;
    const float* ps = S + (size_t)b * CC * NN + n;
    const float* pt = T + (size_t)b * CC * NN + n;
    float as = 0.f, at = 0.f;
#pragma unroll 4
    for (int c = 0; c < CC; ++c) {
        as = fmaf(ps[(size_t)c * NN], wm_s[c], as);
        at = fmaf(pt[(size_t)c * NN], wm_t[c], at);
    }
    mask[(0 * BB + b) * NN + n] = as + bm_s[0];
    mask[(1 * BB + b) * NN + n] = at + bm_t[0];
}

// ---------------------------------------------------------------------------
// Kernel 2: softmax over N, rewrite mask in-place as attention weights.
// One 256-thread block per (st,b).
// ---------------------------------------------------------------------------
__global__ void k_softmax(float* __restrict__ mask, float* __restrict__ stats) {
    const int sb = blockIdx.x;           // st*BB + b
    float* m = mask + (size_t)sb * NN;
    __shared__ float red[256];
    const int tid = threadIdx.x;

    float mx = -3.402823466e38f;
    for (int i = tid; i < NN; i += 256) mx = fmaxf(mx, m[i]);
    red[tid] = mx; __syncthreads();
    for (int s = 128; s > 0; s >>= 1) {
        if (tid < s) red[tid] = fmaxf(red[tid], red[tid + s]);
        __syncthreads();
    }
    mx = red[0]; __syncthreads();

    float sum = 0.f;
    for (int i = tid; i < NN; i += 256) sum += expf(m[i] - mx);
    red[tid] = sum; __syncthreads();
    for (int s = 128; s > 0; s >>= 1) {
        if (tid < s) red[tid] += red[tid + s];
        __syncthreads();
    }
    sum = red[0];
    const float inv = 1.f / sum;
    for (int i = tid; i < NN; i += 256) m[i] = expf(m[i] - mx) * inv;
    if (tid == 0) { stats[sb * 2] = mx; stats[sb * 2 + 1] = sum; }
}

// ---------------------------------------------------------------------------
// Kernel 3: ctx[st][b][c] = sum_n x[b][c][n] * attn[st][b][n]
// One 256-thread block per (c, b, st); float4 streaming.
// ---------------------------------------------------------------------------
__global__ void k_ctx(const float* __restrict__ S, const float* __restrict__ T,
                      const float* __restrict__ mask, float* __restrict__ ctx) {
    const int c = blockIdx.x, b = blockIdx.y, st = blockIdx.z;
    const float* x = (st ? T : S) + ((size_t)(b * CC) + c) * NN;
    const float* at = mask + ((size_t)st * BB + b) * NN;
    const float4* x4 = (const float4*)x;
    const float4* a4 = (const float4*)at;
    const int tid = threadIdx.x;
    float acc = 0.f;
    for (int i = tid; i < NN / 4; i += 256) {
        float4 xv = x4[i], av = a4[i];
        acc = fmaf(xv.x, av.x, acc);
        acc = fmaf(xv.y, av.y, acc);
        acc = fmaf(xv.z, av.z, acc);
        acc = fmaf(xv.w, av.w, acc);
    }
    __shared__ float red[256];
    red[tid] = acc; __syncthreads();
    for (int s = 128; s > 0; s >>= 1) {
        if (tid < s) red[tid] += red[tid + s];
        __syncthreads();
    }
    if (tid == 0) ctx[((size_t)st * BB + b) * CC + c] = red[0];
}

// ---------------------------------------------------------------------------
// Kernel 4: tiny MLP with fp32 WMMA (V_WMMA_F32_16X16X4_F32), single block,
// 256 threads = 8 wave32 waves. A-matrix rows padded to 16 with zeros so
// EXEC stays all-1s at every WMMA.
// ---------------------------------------------------------------------------
__global__ void k_mlp(const float* __restrict__ ctx,
                      const float* __restrict__ w1_s, const float* __restrict__ b1_s,
                      const float* __restrict__ g_s,  const float* __restrict__ be_s,
                      const float* __restrict__ w2_s, const float* __restrict__ b2_s,
                      const float* __restrict__ w1_t, const float* __restrict__ b1_t,
                      const float* __restrict__ g_t,  const float* __restrict__ be_t,
                      const float* __restrict__ w2_t, const float* __restrict__ b2_t,
                      float* __restrict__ add) {
    __shared__ float sA[16 * CC];     // ctx padded: rows 0..1 real, 2..15 zero
    __shared__ float sHn[16 * CHC];   // hidden padded the same way
    __shared__ float sStat[4];        // sum[b], sumsq[b]

    const int tid = threadIdx.x;
    const int wave = tid >> 5, lane = tid & 31;
    const int half = lane >> 4, p = lane & 15;

    for (int i = tid; i < 16 * CC; i += 256) sA[i] = 0.f;
    for (int i = tid; i < 16 * CHC; i += 256) sHn[i] = 0.f;

    for (int st = 0; st < 2; ++st) {
        const float* w1 = st ? w1_t : w1_s;
        const float* b1 = st ? b1_t : b1_s;
        const float* g  = st ? g_t  : g_s;
        const float* be = st ? be_t : be_s;
        const float* w2 = st ? w2_t : w2_s;
        const float* b2 = st ? b2_t : b2_s;

        __syncthreads();
        for (int i = tid; i < BB * CC; i += 256) sA[i] = ctx[st * BB * CC + i];
        if (tid < 4) sStat[tid] = 0.f;
        __syncthreads();

        // ---- GEMM1: H[2,128] = CTX[2,256] @ W1^T ; wave owns N-tile `wave`
        {
            v8f acc = {0.f, 0.f, 0.f, 0.f, 0.f, 0.f, 0.f, 0.f};
            const float* w1row = w1 + (size_t)(wave * 16 + p) * CC;
            for (int k0 = 0; k0 < CC; k0 += 4) {
                const int kk = k0 + half * 2;
                v2f a; a.x = sA[p * CC + kk]; a.y = sA[p * CC + kk + 1];
                v2f bmat; bmat.x = w1row[kk]; bmat.y = w1row[kk + 1];
                acc = __builtin_amdgcn_wmma_f32_16x16x4_f32(
                    false, a, false, bmat, (short)0, acc, false, false);
            }
            if (lane < 16) {
                const int ch = wave * 16 + lane;
                sHn[0 * CHC + ch] = acc[0] + b1[ch];   // D row M=0 -> batch 0
                sHn[1 * CHC + ch] = acc[1] + b1[ch];   // D row M=1 -> batch 1
            }
        }
        __syncthreads();

        // ---- LayerNorm over CH per batch + ReLU (rows 2..15 remain zero)
        {
            const int bi = tid >> 7, chi = tid & 127;
            const float v = sHn[bi * CHC + chi];
            atomicAdd(&sStat[bi], v);
            atomicAdd(&sStat[2 + bi], v * v);
            __syncthreads();
            const float mu = sStat[bi] * (1.f / CHC);
            const float var = sStat[2 + bi] * (1.f / CHC) - mu * mu;
            float hn = (v - mu) * rsqrtf(var + LN_EPS) * g[chi] + be[chi];
            hn = fmaxf(hn, 0.f);
            __syncthreads();
            sHn[bi * CHC + chi] = hn;
        }
        __syncthreads();

        // ---- GEMM2: ADD[2,256] = Hn[2,128] @ W2^T ; wave owns tiles {wave, wave+8}
        for (int t = 0; t < 2; ++t) {
            const int nt = wave + t * 8;
            const float* w2row = w2 + (size_t)(nt * 16 + p) * CHC;
            v8f acc = {0.f, 0.f, 0.f, 0.f, 0.f, 0.f, 0.f, 0.f};
            for (int k0 = 0; k0 < CHC; k0 += 4) {
                const int kk = k0 + half * 2;
                v2f a; a.x = sHn[p * CHC + kk]; a.y = sHn[p * CHC + kk + 1];
                v2f bmat; bmat.x = w2row[kk]; bmat.y = w2row[kk + 1];
                acc = __builtin_amdgcn_wmma_f32_16x16x4_f32(
                    false, a, false, bmat, (short)0, acc, false, false);
            }
            if (lane < 16) {
                const int n = nt * 16 + lane;
                add[st * BB * CC + 0 * CC + n] = acc[0] + b2[n];
                add[st * BB * CC + 1 * CC + n] = acc[1] + b2[n];
            }
        }
        __syncthreads();
    }
}

// ---------------------------------------------------------------------------
// Kernel 5: loss = (1/B) * sum ((S - T) + (add_s - add_t))^2
// ---------------------------------------------------------------------------
__global__ void k_zero(float* out) { out[0] = 0.f; }

__global__ void k_loss(const float* __restrict__ S, const float* __restrict__ T,
                       const float* __restrict__ add, float* __restrict__ out) {
    const int chunk = blockIdx.x;      // 4 chunks of 8192 per row
    const int c = blockIdx.y, b = blockIdx.z;
    const size_t base = ((size_t)(b * CC) + c) * NN + (size_t)chunk * 8192;
    const float ad = add[0 * BB * CC + b * CC + c] - add[1 * BB * CC + b * CC + c];
    const float4* s4 = (const float4*)(S + base);
    const float4* t4 = (const float4*)(T + base);
    const int tid = threadIdx.x;
    float acc = 0.f;
    for (int i = tid; i < 2048; i += 256) {
        if (i + 256 < 2048) {
            __builtin_prefetch((const void*)(s4 + i + 256), 0, 0);
            __builtin_prefetch((const void*)(t4 + i + 256), 0, 0);
        }
        const float4 sv = s4[i], tv = t4[i];
        float d0 = (sv.x - tv.x) + ad;
        float d1 = (sv.y - tv.y) + ad;
        float d2 = (sv.z - tv.z) + ad;
        float d3 = (sv.w - tv.w) + ad;
        acc = fmaf(d0, d0, acc);
        acc = fmaf(d1, d1, acc);
        acc = fmaf(d2, d2, acc);
        acc = fmaf(d3, d3, acc);
    }
    __shared__ float red[256];
    red[tid] = acc; __syncthreads();
    for (int s = 128; s > 0; s >>= 1) {
        if (tid < s) red[tid] += red[tid + s];
        __syncthreads();
    }
    if (tid == 0) atomicAdd(out, red[0] * (1.f / BB));
}

// ---------------------------------------------------------------------------
extern "C" void kernel_launch(void* const* d_in, const int* in_sizes, int n_in,
                              void* d_out, int out_size, void* d_ws, size_t ws_size,
                              hipStream_t stream) {
    const float* S    = (const float*)d_in[0];
    const float* T    = (const float*)d_in[1];
    const float* wm_s = (const float*)d_in[2];
    const float* bm_s = (const float*)d_in[3];
    const float* wm_t = (const float*)d_in[4];
    const float* bm_t = (const float*)d_in[5];
    const float* w1_s = (const float*)d_in[6];
    const float* b1_s = (const float*)d_in[7];
    const float* g_s  = (const float*)d_in[8];
    const float* be_s = (const float*)d_in[9];
    const float* w2_s = (const float*)d_in[10];
    const float* b2_s = (const float*)d_in[11];
    const float* w1_t = (const float*)d_in[12];
    const float* b1_t = (const float*)d_in[13];
    const float* g_t  = (const float*)d_in[14];
    const float* be_t = (const float*)d_in[15];
    const float* w2_t = (const float*)d_in[16];
    const float* b2_t = (const float*)d_in[17];
    float* out = (float*)d_out;

    float* ws    = (float*)d_ws;
    float* mask  = ws;                       // [2][B][N]  = 131072 floats
    float* stats = mask + 2 * BB * NN;       // 8 floats
    float* ctx   = stats + 8;                // [2][B][C]  = 1024 floats
    float* add   = ctx + 2 * BB * CC;        // [2][B][C]  = 1024 floats

    k_mask<<<dim3(NN / 256, BB), 256, 0, stream>>>(S, T, wm_s, bm_s, wm_t, bm_t, mask);
    k_softmax<<<4, 256, 0, stream>>>(mask, stats);
    k_ctx<<<dim3(CC, BB, 2), 256, 0, stream>>>(S, T, mask, ctx);
    k_mlp<<<1, 256, 0, stream>>>(ctx, w1_s, b1_s, g_s, be_s, w2_s, b2_s,
                                 w1_t, b1_t, g_t, be_t, w2_t, b2_t, add);
    k_zero<<<1, 1, 0, stream>>>(out);
    k_loss<<<dim3(4, CC, BB), 256, 0, stream>>>(S, T, add, out);
}